// DPPLayer_2688649527915
// MI455X (gfx1250) — compile-verified
//
#include <hip/hip_runtime.h>
#include <hip/hip_bf16.h>

// ---------------------------------------------------------------------------
// DPP sampling from SVD of a 131072x64 f32 matrix, CDNA5 (gfx1250) version.
//   G = E^T E            (WMMA f32 16x16x4, atomics into 64x64)
//   Jacobi eigensolve    (single WGP, LDS)
//   V0 = E * (Veig*sel/s)(WMMA GEMM)
//   64x { gumbel-argmax choice, rank-1 elimination, Cholesky-QR re-orth }
// All data-dependent iteration control lives in device memory (graph-safe).
// ---------------------------------------------------------------------------

#define N_ROWS 131072
#define D_COLS 64
#define GRAM_ROWS 256

typedef __attribute__((ext_vector_type(2))) float v2f;
typedef __attribute__((ext_vector_type(8))) float v8f;

#define CTL_K      0
#define CTL_J      1
#define CTL_ACTIVE 8   // 64 entries

__device__ __forceinline__ unsigned hash_u32(unsigned x) {
  x ^= x >> 16; x *= 0x7feb352dU;
  x ^= x >> 15; x *= 0x846ca68bU;
  x ^= x >> 16; return x;
}
__device__ __forceinline__ float u01(unsigned h) {
  return ((h >> 8) + 0.5f) * (1.0f / 16777216.0f);
}

// ---------------------------------------------------------------------------
// Zero output subset and Gram accumulator.
__global__ void k_zero(float* __restrict__ subset, float* __restrict__ G) {
  long i = (long)blockIdx.x * blockDim.x + threadIdx.x;
  if (i < N_ROWS) subset[i] = 0.0f;
  if (i < 64 * 64) G[i] = 0.0f;
}

// ---------------------------------------------------------------------------
// G += src^T src over a 256-row slab. 128 threads = 4 waves; wave w owns the
// 16-row block a0=w*16 of G and sweeps all four 16-col blocks.
// WMMA f32 16x16x4: A[M][k] = src[n0+k][a0+M], B[k][n] = src[n0+k][j0+n].
__global__ void k_gram(const float* __restrict__ src, float* __restrict__ G,
                       const unsigned* __restrict__ ctl, int iter) {
  if (iter >= 0 && iter >= (int)ctl[CTL_K]) return;
  const int wave = threadIdx.x >> 5;
  const int lane = threadIdx.x & 31;
  const int M  = lane & 15;
  const int kb = (lane >> 4) << 1;        // 0 or 2
  const int a0 = wave * 16;
  const long n0 = (long)blockIdx.x * GRAM_ROWS;

  v8f acc0 = {}, acc1 = {}, acc2 = {}, acc3 = {};
  for (int kk = 0; kk < GRAM_ROWS; kk += 4) {
    const long n = n0 + kk + kb;
    v2f a;
    a.x = src[n * 64 + a0 + M];
    a.y = src[(n + 1) * 64 + a0 + M];
    v2f b0, b1, b2, b3;
    b0.x = src[n * 64 +  0 + M]; b0.y = src[(n + 1) * 64 +  0 + M];
    b1.x = src[n * 64 + 16 + M]; b1.y = src[(n + 1) * 64 + 16 + M];
    b2.x = src[n * 64 + 32 + M]; b2.y = src[(n + 1) * 64 + 32 + M];
    b3.x = src[n * 64 + 48 + M]; b3.y = src[(n + 1) * 64 + 48 + M];
    acc0 = __builtin_amdgcn_wmma_f32_16x16x4_f32(false, a, false, b0, (short)0, acc0, false, false);
    acc1 = __builtin_amdgcn_wmma_f32_16x16x4_f32(false, a, false, b1, (short)0, acc1, false, false);
    acc2 = __builtin_amdgcn_wmma_f32_16x16x4_f32(false, a, false, b2, (short)0, acc2, false, false);
    acc3 = __builtin_amdgcn_wmma_f32_16x16x4_f32(false, a, false, b3, (short)0, acc3, false, false);
  }
  const int mhi = (lane >> 4) << 3;       // +8 for lanes 16..31
  for (int v = 0; v < 8; ++v) {
    const int row = a0 + v + mhi;
    atomicAdd(&G[row * 64 +  0 + M], acc0[v]);
    atomicAdd(&G[row * 64 + 16 + M], acc1[v]);
    atomicAdd(&G[row * 64 + 32 + M], acc2[v]);
    atomicAdd(&G[row * 64 + 48 + M], acc3[v]);
  }
}

// ---------------------------------------------------------------------------
// dst[r0:r0+16, :] = src[r0:r0+16, :] @ Wm  (64x64). LDS-staged so src==dst
// (in-place re-orthonormalization) is safe. 128 threads = 4 waves, wave w
// produces output column block j0=w*16.
__global__ void k_gemm(const float* __restrict__ src, float* __restrict__ dst,
                       const float* __restrict__ Wm,
                       const unsigned* __restrict__ ctl, int iter) {
  if (iter >= 0 && iter >= (int)ctl[CTL_K]) return;
  __shared__ float tile[16][64];
  const long r0 = (long)blockIdx.x * 16;
  for (int idx = threadIdx.x; idx < 16 * 64; idx += blockDim.x) {
    const int r = idx >> 6, c = idx & 63;
    tile[r][c] = src[(r0 + r) * 64 + c];
  }
  __syncthreads();
  const int wave = threadIdx.x >> 5;
  const int lane = threadIdx.x & 31;
  const int M  = lane & 15;
  const int kb = (lane >> 4) << 1;
  const int j0 = wave * 16;
  v8f acc = {};
  for (int k = 0; k < 64; k += 4) {
    v2f a;
    a.x = tile[M][k + kb];
    a.y = tile[M][k + kb + 1];
    v2f b;
    b.x = Wm[(k + kb) * 64 + j0 + M];
    b.y = Wm[(k + kb + 1) * 64 + j0 + M];
    acc = __builtin_amdgcn_wmma_f32_16x16x4_f32(false, a, false, b, (short)0, acc, false, false);
  }
  const int mhi = (lane >> 4) << 3;
  for (int v = 0; v < 8; ++v)
    dst[(r0 + v + mhi) * 64 + j0 + M] = acc[v];
}

// ---------------------------------------------------------------------------
// Jacobi eigensolve of G (64x64, LDS), sort desc, Bernoulli e/(e+1) column
// selection, build W0 = Veig * diag(sel/s). One workgroup of 64 threads.
__global__ void k_eig(const float* __restrict__ G, float* __restrict__ Wm,
                      unsigned* __restrict__ ctl,
                      unsigned long long* __restrict__ amax) {
  __shared__ float A[64][64];
  __shared__ float Vv[64][64];
  __shared__ float cs[2];
  __shared__ float ev[64];
  __shared__ int ord[64];
  __shared__ int selv[64];
  const int t = threadIdx.x;
  for (int c = 0; c < 64; ++c) {
    A[t][c]  = G[t * 64 + c];
    Vv[t][c] = (t == c) ? 1.0f : 0.0f;
  }
  __syncthreads();
  for (int sweep = 0; sweep < 12; ++sweep) {
    for (int p = 0; p < 63; ++p) {
      for (int q = p + 1; q < 64; ++q) {
        if (t == 0) {
          const float apq = A[p][q];
          float c_ = 1.0f, s_ = 0.0f;
          if (fabsf(apq) > 1e-12f) {
            const float tau = (A[q][q] - A[p][p]) / (2.0f * apq);
            const float tt = (tau >= 0.0f ? 1.0f : -1.0f) /
                             (fabsf(tau) + sqrtf(1.0f + tau * tau));
            c_ = rsqrtf(1.0f + tt * tt);
            s_ = tt * c_;
          }
          cs[0] = c_; cs[1] = s_;
        }
        __syncthreads();
        const float c_ = cs[0], s_ = cs[1];
        if (s_ != 0.0f) {                         // uniform branch
          const float ap = A[t][p], aq = A[t][q];
          A[t][p] = c_ * ap - s_ * aq;
          A[t][q] = s_ * ap + c_ * aq;
          const float vp = Vv[t][p], vq = Vv[t][q];
          Vv[t][p] = c_ * vp - s_ * vq;
          Vv[t][q] = s_ * vp + c_ * vq;
          __syncthreads();
          const float rp = A[p][t], rq = A[q][t];
          A[p][t] = c_ * rp - s_ * rq;
          A[q][t] = s_ * rp + c_ * rq;
        }
        __syncthreads();
      }
    }
  }
  if (t == 0) {
    for (int i = 0; i < 64; ++i) { ev[i] = A[i][i]; ord[i] = i; }
    for (int i = 0; i < 63; ++i) {          // selection sort, descending
      int best = i;
      for (int j = i + 1; j < 64; ++j) if (ev[j] > ev[best]) best = j;
      float te = ev[i]; ev[i] = ev[best]; ev[best] = te;
      int   to = ord[i]; ord[i] = ord[best]; ord[best] = to;
    }
  }
  __syncthreads();
  const float lam = fmaxf(ev[t], 0.0f);
  const float sv  = sqrtf(lam);
  const float pr  = lam / (lam + 1.0f);
  const unsigned h = hash_u32(0xC0FFEE42u + (unsigned)t * 0x9E3779B9u);
  const int sel = (u01(h) < pr) ? 1 : 0;
  selv[t] = sel;
  ctl[CTL_ACTIVE + t] = (unsigned)sel;
  __syncthreads();
  if (t == 0) {
    int k = 0;
    for (int i = 0; i < 64; ++i) k += selv[i];
    ctl[CTL_K] = (unsigned)k;
    ctl[CTL_J] = 0u;
    *amax = 0ULL;
  }
  const float inv_s = sel ? (1.0f / fmaxf(sv, 1e-20f)) : 0.0f;
  const int oc = ord[t];
  for (int r = 0; r < 64; ++r)
    Wm[r * 64 + t] = Vv[r][oc] * inv_s;   // column t of W0
}

// ---------------------------------------------------------------------------
// Gumbel-max categorical over row norms: key = log(max(|V_r|^2,eps)) + g_r.
__global__ void k_choice(const float* __restrict__ V,
                         unsigned long long* __restrict__ amax,
                         const unsigned* __restrict__ ctl, int iter) {
  if (iter >= (int)ctl[CTL_K]) return;
  const long r = (long)blockIdx.x * blockDim.x + threadIdx.x;
  float p = 0.0f;
  for (int c = 0; c < 64; ++c) {
    const float v = V[r * 64 + c];
    p += v * v;
  }
  const float lp = logf(fmaxf(p, 1e-30f));
  const unsigned seed = hash_u32(0x51E2D1u + (unsigned)iter * 0x9E3779B9u);
  const unsigned h = hash_u32(seed ^ (unsigned)r);
  const float u = u01(h);
  const float g = -logf(-logf(u));
  const float key = lp + g;
  unsigned kb = __float_as_uint(key);
  kb = (kb & 0x80000000u) ? ~kb : (kb | 0x80000000u);   // monotone map
  const unsigned long long packed = ((unsigned long long)kb << 32) | (unsigned)r;
  atomicMax(amax, packed);
}

// ---------------------------------------------------------------------------
// Finalize the choice: subset[choice]=1, j = argmax_c |row_c|*active_c,
// scale_c = row_c/row_j, deactivate j, zero G for the upcoming Gram, reset
// the argmax cell. One workgroup of 64 threads.
__global__ void k_pick(const float* __restrict__ V, float* __restrict__ subset,
                       float* __restrict__ G, float* __restrict__ scale,
                       unsigned* __restrict__ ctl,
                       unsigned long long* __restrict__ amax, int iter) {
  if (iter >= (int)ctl[CTL_K]) return;
  __shared__ float srow[64];
  __shared__ float sabs[64];
  const int t = threadIdx.x;
  const unsigned choice = (unsigned)((*amax) & 0xffffffffu);
  const float rv = V[(long)choice * 64 + t];
  srow[t] = rv;
  sabs[t] = fabsf(rv) * (ctl[CTL_ACTIVE + t] ? 1.0f : 0.0f);
  __syncthreads();
  if (t == 0) {
    int j = 0; float best = sabs[0];
    for (int c = 1; c < 64; ++c)
      if (sabs[c] > best) { best = sabs[c]; j = c; }
    ctl[CTL_J] = (unsigned)j;
    subset[choice] = 1.0f;
    *amax = 0ULL;
  }
  __syncthreads();
  const int j = (int)ctl[CTL_J];
  const float denom = srow[j];
  scale[t] = (denom != 0.0f) ? (srow[t] / denom) : 0.0f;
  if (t == 0) ctl[CTL_ACTIVE + j] = 0u;
  for (int idx = t; idx < 64 * 64; idx += 64) G[idx] = 0.0f;
}

// ---------------------------------------------------------------------------
// Rank-1 elimination: V[r,:] -= V[r,j]*scale[:], then mask by active.
__global__ void k_rank1(float* __restrict__ V, const float* __restrict__ scale,
                        const unsigned* __restrict__ ctl, int iter) {
  if (iter >= (int)ctl[CTL_K]) return;
  const long r = (long)blockIdx.x * blockDim.x + threadIdx.x;
  const int j = (int)ctl[CTL_J];
  const float vj = V[r * 64 + j];
  for (int c = 0; c < 64; ++c) {
    const float nv = V[r * 64 + c] - vj * scale[c];
    V[r * 64 + c] = ctl[CTL_ACTIVE + c] ? nv : 0.0f;
  }
}

// ---------------------------------------------------------------------------
// Cholesky-QR factor: masked Cholesky of G = V^T V, W = L^{-T} with dead
// (inactive or tiny-pivot) columns zeroed -> V@W is orthonormalized like the
// reference's Gram-Schmidt with its nrm>1e-10 guard. 64 threads.
__global__ void k_orth(const float* __restrict__ G, float* __restrict__ Wm,
                       const unsigned* __restrict__ ctl, int iter) {
  if (iter >= (int)ctl[CTL_K]) return;
  __shared__ float L[64][64];
  __shared__ float Li[64][64];
  __shared__ int dead[64];
  const int t = threadIdx.x;
  for (int c = 0; c < 64; ++c) L[t][c] = G[t * 64 + c];
  __syncthreads();
  for (int c = 0; c < 64; ++c) {
    if (t == c) {
      float d = L[c][c];
      for (int k = 0; k < c; ++k) d -= L[c][k] * L[c][k];
      if (!ctl[CTL_ACTIVE + c] || d <= 1e-20f) { dead[c] = 1; L[c][c] = 1.0f; }
      else { dead[c] = 0; L[c][c] = sqrtf(d); }
    }
    __syncthreads();
    if (t > c) {
      if (dead[c]) L[t][c] = 0.0f;
      else {
        float sum = L[t][c];
        for (int k = 0; k < c; ++k) sum -= L[t][k] * L[c][k];
        L[t][c] = sum / L[c][c];
      }
    }
    __syncthreads();
  }
  // Column t of Li = L^{-1} (lower triangular), dead rows zeroed.
  for (int i = 0; i < t; ++i) Li[i][t] = 0.0f;
  Li[t][t] = dead[t] ? 0.0f : 1.0f / L[t][t];
  for (int i = t + 1; i < 64; ++i) {
    float sum = 0.0f;
    for (int k = t; k < i; ++k) sum += L[i][k] * Li[k][t];
    Li[i][t] = dead[i] ? 0.0f : (-sum / L[i][i]);
  }
  __syncthreads();
  for (int c = 0; c < 64; ++c)
    Wm[t * 64 + c] = dead[c] ? 0.0f : Li[c][t];   // W = L^{-T}, masked
}

// ---------------------------------------------------------------------------
extern "C" void kernel_launch(void* const* d_in, const int* in_sizes, int n_in,
                              void* d_out, int out_size, void* d_ws, size_t ws_size,
                              hipStream_t stream) {
  (void)in_sizes; (void)n_in; (void)out_size; (void)ws_size;
  const float* E = (const float*)d_in[0];
  float* subset = (float*)d_out;

  float* ws   = (float*)d_ws;
  float* G    = ws;                         // 4096 f32
  float* Wm   = ws + 4096;                  // 4096 f32
  float* scale = ws + 8192;                 // 64 f32
  unsigned* ctl = (unsigned*)(ws + 8320);   // 128 u32
  unsigned long long* amax = (unsigned long long*)(ws + 8448); // 8B aligned
  float* V    = ws + 8704;                  // N_ROWS*64 f32 (~33.5 MB)

  const int r = D_COLS;

  k_zero<<<(N_ROWS + 255) / 256, 256, 0, stream>>>(subset, G);
  k_gram<<<N_ROWS / GRAM_ROWS, 128, 0, stream>>>(E, G, ctl, -1);
  k_eig<<<1, 64, 0, stream>>>(G, Wm, ctl, amax);
  k_gemm<<<N_ROWS / 16, 128, 0, stream>>>(E, V, Wm, ctl, -1);

  for (int i = 0; i < r; ++i) {
    k_choice<<<N_ROWS / 256, 256, 0, stream>>>(V, amax, ctl, i);
    k_pick<<<1, 64, 0, stream>>>(V, subset, G, scale, ctl, amax, i);
    k_rank1<<<N_ROWS / 256, 256, 0, stream>>>(V, scale, ctl, i);
    k_gram<<<N_ROWS / GRAM_ROWS, 128, 0, stream>>>(V, G, ctl, i);
    k_orth<<<1, 64, 0, stream>>>(G, Wm, ctl, i);
    k_gemm<<<N_ROWS / 16, 128, 0, stream>>>(V, V, Wm, ctl, i);
  }
}